// GATLayer_37469294691136
// MI455X (gfx1250) — compile-verified
//
#include <hip/hip_runtime.h>
#include <hip/hip_bf16.h>

#define IN_DIM  256
#define OUT_DIM 256
#define NHEAD   4
#define HDIM    64
#define SLOPE_ATTN 0.2f
#define SLOPE_OUT  0.01f

typedef __attribute__((ext_vector_type(2))) float v2f;
typedef __attribute__((ext_vector_type(8))) float v8f;

// ---------------------------------------------------------------------------
// K0: zero the output accumulator, init emax/denom
// ---------------------------------------------------------------------------
__global__ void __launch_bounds__(256)
init_kernel(float* __restrict__ out, int outN,
            float* __restrict__ emax, float* __restrict__ denom, int NH) {
    int i = blockIdx.x * 256 + threadIdx.x;
    if (i < outN) out[i] = 0.0f;
    if (i < NH) { emax[i] = -1.0e30f; denom[i] = 0.0f; }
}

// ---------------------------------------------------------------------------
// K1: feat[N,256] = h[N,256] @ W[256,256] via V_WMMA_F32_16X16X4_F32.
// Block = 256 threads = 8 waves. Each wave computes a 16x128 output tile
// (8 v8f accumulators). W k-slices (32x128) staged in LDS for reuse.
// ---------------------------------------------------------------------------
__global__ void __launch_bounds__(256)
gemm_feat_kernel(const float* __restrict__ h, const float* __restrict__ W,
                 float* __restrict__ feat, int Nnodes) {
    __shared__ float ldsW[32 * 128];            // 16 KB

    const int tid  = threadIdx.x;
    const int wave = tid >> 5;
    const int lane = tid & 31;
    const int mtile  = blockIdx.x * 8 + wave;   // 16-row strip index
    const int mtiles = (Nnodes + 15) >> 4;
    const int n0   = blockIdx.y * 128;
    const bool active = (mtile < mtiles);

    // A-frag addressing: lane L holds row M = L%16; lanes 0-15 carry K=k0,k0+1
    // in (a.x,a.y), lanes 16-31 carry K=k0+2,k0+3.
    int m = mtile * 16 + (lane & 15);
    if (m >= Nnodes) m = Nnodes - 1;            // clamp (values unused OOB)
    const int kl = (lane >> 4) * 2;             // 0 or 2
    const int nl = lane & 15;
    const float* hrow = h + (size_t)m * IN_DIM;

    v8f acc[8] = {};

    for (int k0 = 0; k0 < IN_DIM; k0 += 32) {
        __syncthreads();                        // protect previous LDS reads
        // stage W[k0:k0+32, n0:n0+128] -> LDS (coalesced, 16 floats/thread)
        #pragma unroll
        for (int i = 0; i < 16; ++i) {
            int idx = tid + 256 * i;            // 0..4095
            int r = idx >> 7, c = idx & 127;
            ldsW[idx] = W[(size_t)(k0 + r) * OUT_DIM + n0 + c];
        }
        __syncthreads();

        if (active) {
            #pragma unroll
            for (int kk = 0; kk < 32; kk += 4) {
                v2f a = *(const v2f*)(hrow + k0 + kk + kl);
                #pragma unroll
                for (int j = 0; j < 8; ++j) {
                    int base = (kk + kl) * 128 + j * 16 + nl;
                    v2f b;
                    b.x = ldsW[base];
                    b.y = ldsW[base + 128];
                    acc[j] = __builtin_amdgcn_wmma_f32_16x16x4_f32(
                        false, a, false, b, (short)0, acc[j], false, false);
                }
            }
        }
    }

    if (active) {
        const int rowhi = (lane >> 4) * 8;      // C/D: VGPR r -> M=r (+8 hi half)
        #pragma unroll
        for (int j = 0; j < 8; ++j) {
            #pragma unroll
            for (int r = 0; r < 8; ++r) {
                int row = mtile * 16 + rowhi + r;
                if (row < Nnodes)
                    feat[(size_t)row * OUT_DIM + n0 + j * 16 + nl] = acc[j][r];
            }
        }
    }
}

// ---------------------------------------------------------------------------
// K2: el[n,h] = <feat[n,h,:], attn_l[h,:]>, er likewise. One wave per (n,h).
// ---------------------------------------------------------------------------
__global__ void __launch_bounds__(256)
elr_kernel(const float* __restrict__ feat, const float* __restrict__ attn_l,
           const float* __restrict__ attn_r, float* __restrict__ el,
           float* __restrict__ er, int NH) {
    int w    = (blockIdx.x * 256 + threadIdx.x) >> 5;
    int lane = threadIdx.x & 31;
    if (w >= NH) return;                        // wave-uniform
    int n = w >> 2, hh = w & 3;
    const float* f  = feat + (size_t)n * OUT_DIM + hh * HDIM;
    const float* al = attn_l + hh * HDIM;
    const float* ar = attn_r + hh * HDIM;
    float sl = f[lane] * al[lane] + f[lane + 32] * al[lane + 32];
    float sr = f[lane] * ar[lane] + f[lane + 32] * ar[lane + 32];
    #pragma unroll
    for (int off = 16; off > 0; off >>= 1) {
        sl += __shfl_down(sl, off);
        sr += __shfl_down(sr, off);
    }
    if (lane == 0) { el[w] = sl; er[w] = sr; }
}

// ---------------------------------------------------------------------------
// sign-aware float atomic max via integer atomics (monotone bit mapping)
// ---------------------------------------------------------------------------
__device__ __forceinline__ void atomicMaxF(float* addr, float v) {
    if (v >= 0.0f) atomicMax((int*)addr, __float_as_int(v));
    else           atomicMin((unsigned int*)addr, (unsigned int)__float_as_int(v));
}

__device__ __forceinline__ float leaky(float x, float s) {
    return x > 0.0f ? x : x * s;
}

// ---------------------------------------------------------------------------
// K3: per-(edge,head) score -> segment max into emax[dst,h]
// ---------------------------------------------------------------------------
__global__ void __launch_bounds__(256)
edge_max_kernel(const int* __restrict__ src, const int* __restrict__ dst,
                const float* __restrict__ el, const float* __restrict__ er,
                float* __restrict__ emax, int E4) {
    int i = blockIdx.x * 256 + threadIdx.x;
    if (i >= E4) return;
    int e = i >> 2, hh = i & 3;
    int s = src[e], d = dst[e];
    float sc = leaky(el[s * 4 + hh] + er[d * 4 + hh], SLOPE_ATTN);
    atomicMaxF(&emax[d * 4 + hh], sc);
}

// ---------------------------------------------------------------------------
// K4: recompute score, exp(score - emax), accumulate denom[dst,h]
// ---------------------------------------------------------------------------
__global__ void __launch_bounds__(256)
edge_sum_kernel(const int* __restrict__ src, const int* __restrict__ dst,
                const float* __restrict__ el, const float* __restrict__ er,
                const float* __restrict__ emax, float* __restrict__ denom, int E4) {
    int i = blockIdx.x * 256 + threadIdx.x;
    if (i >= E4) return;
    int e = i >> 2, hh = i & 3;
    int s = src[e], d = dst[e];
    float sc = leaky(el[s * 4 + hh] + er[d * 4 + hh], SLOPE_ATTN);
    float ex = __expf(sc - emax[d * 4 + hh]);
    atomicAdd(&denom[d * 4 + hh], ex);
}

// ---------------------------------------------------------------------------
// K5: wave per edge. Lanes 0-3 compute per-head softmax weight a[h],
// broadcast via shfl; 8 coalesced 128B atomic-add sweeps into out[dst].
// ---------------------------------------------------------------------------
__global__ void __launch_bounds__(256)
edge_agg_kernel(const int* __restrict__ src, const int* __restrict__ dst,
                const float* __restrict__ feat, const float* __restrict__ el,
                const float* __restrict__ er, const float* __restrict__ emax,
                const float* __restrict__ denom, float* __restrict__ out, int E) {
    int w    = (blockIdx.x * 256 + threadIdx.x) >> 5;
    int lane = threadIdx.x & 31;
    if (w >= E) return;                         // wave-uniform
    int s = src[w], d = dst[w];

    float aval = 0.0f;
    if (lane < 4) {
        float sc = leaky(el[s * 4 + lane] + er[d * 4 + lane], SLOPE_ATTN);
        float ex = __expf(sc - emax[d * 4 + lane]);
        aval = ex / fmaxf(denom[d * 4 + lane], 1e-9f);
    }

    const float* fs = feat + (size_t)s * OUT_DIM;
    float* od = out + (size_t)d * OUT_DIM;
    #pragma unroll
    for (int j = 0; j < 8; ++j) {               // head h = j>>1
        float a_h = __shfl(aval, j >> 1);
        int k = j * 32 + lane;                  // contiguous 128B per sweep
        atomicAdd(&od[k], fs[k] * a_h);
    }
}

// ---------------------------------------------------------------------------
// K6: out = leaky_relu(out + bias, 0.01) in place
// ---------------------------------------------------------------------------
__global__ void __launch_bounds__(256)
finalize_kernel(float* __restrict__ out, const float* __restrict__ bias, int total) {
    int i = blockIdx.x * 256 + threadIdx.x;
    if (i >= total) return;
    float v = out[i] + bias[i & (OUT_DIM - 1)];
    out[i] = leaky(v, SLOPE_OUT);
}

// ---------------------------------------------------------------------------
extern "C" void kernel_launch(void* const* d_in, const int* in_sizes, int n_in,
                              void* d_out, int out_size, void* d_ws, size_t ws_size,
                              hipStream_t stream) {
    const float* h      = (const float*)d_in[0];
    const int*   src    = (const int*)  d_in[1];
    const int*   dst    = (const int*)  d_in[2];
    const float* W      = (const float*)d_in[3];
    const float* attn_l = (const float*)d_in[4];
    const float* attn_r = (const float*)d_in[5];
    const float* bias   = (const float*)d_in[6];
    float* out = (float*)d_out;

    const int N  = in_sizes[0] / IN_DIM;
    const int E  = in_sizes[1];
    const int NH = N * NHEAD;

    // workspace layout (floats): feat[N*256] | el[NH] | er[NH] | emax[NH] | denom[NH]
    float* feat  = (float*)d_ws;
    float* el    = feat + (size_t)N * OUT_DIM;
    float* er    = el + NH;
    float* emax  = er + NH;
    float* denom = emax + NH;

    // K0: init accumulators
    {
        int total = out_size;                   // >= NH
        init_kernel<<<(total + 255) / 256, 256, 0, stream>>>(out, out_size, emax, denom, NH);
    }
    // K1: WMMA GEMM
    {
        int mtiles = (N + 15) / 16;
        dim3 grid((mtiles + 7) / 8, OUT_DIM / 128);
        gemm_feat_kernel<<<grid, 256, 0, stream>>>(h, W, feat, N);
    }
    // K2: el / er
    elr_kernel<<<(NH + 7) / 8, 256, 0, stream>>>(feat, attn_l, attn_r, el, er, NH);
    // K3: segment max
    edge_max_kernel<<<(E * 4 + 255) / 256, 256, 0, stream>>>(src, dst, el, er, emax, E * 4);
    // K4: segment sum of exp
    edge_sum_kernel<<<(E * 4 + 255) / 256, 256, 0, stream>>>(src, dst, el, er, emax, denom, E * 4);
    // K5: weighted scatter-aggregate
    edge_agg_kernel<<<(E + 7) / 8, 256, 0, stream>>>(src, dst, feat, el, er, emax, denom, out, E);
    // K6: bias + outer leaky relu
    finalize_kernel<<<(out_size + 255) / 256, 256, 0, stream>>>(out, bias, out_size);
}